// TimeSeriesInformer_20444044329782
// MI455X (gfx1250) — compile-verified
//
#include <hip/hip_runtime.h>
#include <hip/hip_bf16.h>
#include <math.h>

// Problem constants (match reference)
#define BB   8
#define SS   1024
#define FFI  64
#define DMD  256
#define HH   8
#define NLL  4
#define DFFF 1024
#define NCC  5
#define DD   32
#define SKK  34

typedef float v2f __attribute__((ext_vector_type(2)));
typedef float v8f __attribute__((ext_vector_type(8)));

static __device__ __forceinline__ v8f wmma_f32(v2f a, v2f b, v8f c) {
    // D = A(16x4 f32) x B(4x16 f32) + C(16x16 f32)
    return __builtin_amdgcn_wmma_f32_16x16x4_f32(
        /*neg_a=*/false, a, /*neg_b=*/false, b,
        /*c_mod=*/(short)0, c, /*reuse_a=*/false, /*reuse_b=*/false);
}

// ---------------------------------------------------------------- PE table
__global__ __launch_bounds__(256)
void pe_kernel(float* __restrict__ pe) {
    int id = blockIdx.x * 256 + threadIdx.x;           // over SS*DMD
    int s = id / DMD, c = id % DMD;
    int p = c >> 1;
    float div = __expf(-(float)(2 * p) * (logf(10000.0f) / (float)DMD));
    float ang = (float)s * div;
    pe[id] = (c & 1) ? cosf(ang) : sinf(ang);
}

// ---------------------------------------------------------------- generic WMMA GEMM
// C[M,N] = A[M,K] @ B[K,N] + bias, with fused epilogues.
// 2x2 register blocking: each wave owns a 32x32 output patch (four 16x16 WMMA
// tiles), reusing each A/B fragment across two WMMAs -> 8 bytes per lane of operand
// traffic per 1024-MAC WMMA. Block = 4 waves = 64x64 macro-tile.
enum { EPI_BIAS = 0, EPI_PE = 1, EPI_RELU = 2, EPI_RES = 3, EPI_HEADS = 4 };

template<int EPI>
__global__ __launch_bounds__(128)
void gemm_f32_wmma(const float* __restrict__ A, const float* __restrict__ Bm,
                   const float* __restrict__ bias, const float* __restrict__ extra,
                   float* __restrict__ C, int M, int N, int K)
{
    const int lane = threadIdx.x;           // 0..31
    const int wave = threadIdx.y;           // 0..3
    const int wm = wave >> 1, wn = wave & 1;
    const int lo = lane & 15;
    const int hi = lane >> 4;
    const int kOff = hi * 2;                // A 16x4 layout: halves hold K pairs
    const int m0 = blockIdx.y * 64 + wm * 32;   // rows [m0,m0+16) and [m0+16,m0+32)
    const int n0 = blockIdx.x * 64 + wn * 32;   // cols [n0,n0+16) and [n0+16,n0+32)
    if (m0 >= M) return;

    const float* Arow0 = A + (size_t)(m0 + lo) * K;
    const float* Arow1 = A + (size_t)(m0 + 16 + lo) * K;
    const int nA = n0 + lo;
    const int nB = n0 + 16 + lo;

    v8f acc[2][2];
#pragma unroll
    for (int i = 0; i < 2; ++i)
#pragma unroll
        for (int j = 0; j < 2; ++j)
            acc[i][j] = (v8f){0.f,0.f,0.f,0.f,0.f,0.f,0.f,0.f};

    for (int k = 0; k < K; k += 4) {
        v2f a0, a1, b0, b1;
        a0[0] = Arow0[k + kOff];
        a0[1] = Arow0[k + kOff + 1];
        a1[0] = Arow1[k + kOff];
        a1[1] = Arow1[k + kOff + 1];
        b0[0] = Bm[(size_t)(k + kOff)     * N + nA];
        b0[1] = Bm[(size_t)(k + kOff + 1) * N + nA];
        b1[0] = Bm[(size_t)(k + kOff)     * N + nB];
        b1[1] = Bm[(size_t)(k + kOff + 1) * N + nB];
        acc[0][0] = wmma_f32(a0, b0, acc[0][0]);
        acc[0][1] = wmma_f32(a0, b1, acc[0][1]);
        acc[1][0] = wmma_f32(a1, b0, acc[1][0]);
        acc[1][1] = wmma_f32(a1, b1, acc[1][1]);
    }

    const float bv0 = bias ? bias[nA] : 0.0f;
    const float bv1 = bias ? bias[nB] : 0.0f;
#pragma unroll
    for (int tm = 0; tm < 2; ++tm) {
#pragma unroll
        for (int tn = 0; tn < 2; ++tn) {
            const int n = n0 + tn * 16 + lo;
            const float bv = tn ? bv1 : bv0;
#pragma unroll
            for (int i = 0; i < 8; ++i) {
                int m = m0 + tm * 16 + i + 8 * hi;  // C/D layout: VGPR i, half +8
                float v = acc[tm][tn][i] + bv;
                if (EPI == EPI_PE)   v += extra[(size_t)(m % SS) * DMD + n];
                if (EPI == EPI_RELU) v = v > 0.f ? v : 0.f;
                if (EPI == EPI_RES)  v += extra[(size_t)m * N + n];
                if (EPI == EPI_HEADS) {
                    int b_ = m / SS, s_ = m % SS, hd = n / DD, d_ = n % DD;
                    C[(((size_t)(b_ * HH + hd) * SS + s_) * DD) + d_] = v;
                } else {
                    C[(size_t)m * N + n] = v;
                }
            }
        }
    }
}

// ---------------------------------------------------------------- ProbSparse probe
// mean_attn[bh,key] += sum_q exp(scale*q.k - rowmax_q)   (division by S is
// monotonic and dropped; topk is invariant). Scores stay in registers.
__global__ __launch_bounds__(256)
void probe_kernel(const float* __restrict__ q, const float* __restrict__ k,
                  float* __restrict__ mean_attn)
{
    const int lane = threadIdx.x;           // 0..31
    const int wave = threadIdx.y;           // 0..7  -> keys [wave*128, +128)
    const int qt   = blockIdx.x;            // 0..63 -> 16 queries
    const int bh   = blockIdx.y;            // 0..63
    const int lo = lane & 15, hi = lane >> 4, kOff = hi * 2;
    const float* qb = q + (size_t)bh * SS * DD;
    const float* kb = k + (size_t)bh * SS * DD;
    const int q0 = qt * 16;
    const float scale = 0.17677669529663687f; // 1/sqrt(32)

    // Preload this lane's A-operand fragments for all 8 K-steps (K=D=32)
    float qr[16];
#pragma unroll
    for (int st = 0; st < 8; ++st) {
        qr[2*st]   = qb[(size_t)(q0 + lo) * DD + st*4 + kOff];
        qr[2*st+1] = qb[(size_t)(q0 + lo) * DD + st*4 + kOff + 1];
    }

    v8f acc[8];
#pragma unroll
    for (int kt = 0; kt < 8; ++kt) {
        const int kbase = wave * 128 + kt * 16;
        v8f c = {0.f,0.f,0.f,0.f,0.f,0.f,0.f,0.f};
#pragma unroll
        for (int st = 0; st < 8; ++st) {
            v2f a, b;
            a[0] = qr[2*st]; a[1] = qr[2*st+1];
            b[0] = kb[(size_t)(kbase + lo) * DD + st*4 + kOff];
            b[1] = kb[(size_t)(kbase + lo) * DD + st*4 + kOff + 1];
            c = wmma_f32(a, b, c);
        }
        acc[kt] = c;
    }

    // Per-query row max: reduce over key tiles (regs), lanes in half (shfl),
    // then waves (LDS).
    float rmax[8];
#pragma unroll
    for (int i = 0; i < 8; ++i) {
        float mv = -3.4e38f;
#pragma unroll
        for (int kt = 0; kt < 8; ++kt) mv = fmaxf(mv, acc[kt][i] * scale);
        for (int mask = 1; mask < 16; mask <<= 1)
            mv = fmaxf(mv, __shfl_xor(mv, mask, 32));
        rmax[i] = mv;
    }
    __shared__ float wmax[8][16];
    __shared__ float rowmax[16];
    if (lo == 0) {
#pragma unroll
        for (int i = 0; i < 8; ++i) wmax[wave][hi * 8 + i] = rmax[i];
    }
    __syncthreads();
    int tid = wave * 32 + lane;
    if (tid < 16) {
        float mv = -3.4e38f;
        for (int w = 0; w < 8; ++w) mv = fmaxf(mv, wmax[w][tid]);
        rowmax[tid] = mv;
    }
    __syncthreads();

    // exp(score - rowmax), accumulate per-key column sums.
#pragma unroll
    for (int kt = 0; kt < 8; ++kt) {
        float ksum = 0.f;
#pragma unroll
        for (int i = 0; i < 8; ++i)
            ksum += __expf(acc[kt][i] * scale - rowmax[i + 8 * hi]);
        ksum += __shfl_xor(ksum, 16, 32);   // fold both halves (same key col)
        if (hi == 0) {
            int key = wave * 128 + kt * 16 + lo;
            atomicAdd(&mean_attn[(size_t)bh * SS + key], ksum);
        }
    }
}

// ---------------------------------------------------------------- top-34 selection
__global__ __launch_bounds__(1024)
void topk_kernel(const float* __restrict__ mean_attn, int* __restrict__ idx)
{
    const int bh = blockIdx.x;
    const int t = threadIdx.x;
    __shared__ float sv[1024];
    __shared__ int   si[1024];
    float v = mean_attn[(size_t)bh * SS + t];
    for (int iter = 0; iter < SKK; ++iter) {
        sv[t] = v; si[t] = t;
        __syncthreads();
        for (int s = 512; s > 0; s >>= 1) {
            if (t < s) {
                float ov = sv[t + s]; int oi = si[t + s];
                if (ov > sv[t] || (ov == sv[t] && oi < si[t])) { sv[t] = ov; si[t] = oi; }
            }
            __syncthreads();
        }
        int win = si[0];
        if (t == 0) idx[bh * SKK + iter] = win;
        if (t == win) v = -3.4e38f;
        __syncthreads();
    }
}

// ---------------------------------------------------------------- sparse attention
// All S queries vs 34 sampled keys (staged in LDS); writes [B,S,DM] layout.
__global__ __launch_bounds__(256)
void sparse_attn_kernel(const float* __restrict__ q, const float* __restrict__ k,
                        const float* __restrict__ v, const int* __restrict__ idx,
                        float* __restrict__ o)
{
    const int bh = blockIdx.y;
    const int b = bh / HH, hd = bh % HH;
    const int s = blockIdx.x * 256 + threadIdx.x;
    __shared__ float ks[SKK * DD];
    __shared__ float vs[SKK * DD];
    for (int i = threadIdx.x; i < SKK * DD; i += 256) {
        int j = i / DD, d = i % DD;
        int key = idx[bh * SKK + j];
        ks[i] = k[((size_t)bh * SS + key) * DD + d];
        vs[i] = v[((size_t)bh * SS + key) * DD + d];
    }
    __syncthreads();

    float qr[DD];
#pragma unroll
    for (int d = 0; d < DD; ++d) qr[d] = q[((size_t)bh * SS + s) * DD + d];

    const float scale = 0.17677669529663687f;
    float sc[SKK];
    float mx = -3.4e38f;
    for (int j = 0; j < SKK; ++j) {
        float dot = 0.f;
#pragma unroll
        for (int d = 0; d < DD; ++d) dot += qr[d] * ks[j * DD + d];
        sc[j] = dot * scale;
        mx = fmaxf(mx, sc[j]);
    }
    float oacc[DD];
#pragma unroll
    for (int d = 0; d < DD; ++d) oacc[d] = 0.f;
    float sum = 0.f;
    for (int j = 0; j < SKK; ++j) {
        float w = __expf(sc[j] - mx);
        sum += w;
#pragma unroll
        for (int d = 0; d < DD; ++d) oacc[d] += w * vs[j * DD + d];
    }
    float inv = 1.0f / sum;
#pragma unroll
    for (int d = 0; d < DD; ++d)
        o[((size_t)b * SS + s) * DMD + hd * DD + d] = oacc[d] * inv;
}

// ---------------------------------------------------------------- layernorm (row=256)
__global__ __launch_bounds__(256)
void ln_kernel(const float* __restrict__ y, const float* __restrict__ g,
               const float* __restrict__ be, float* __restrict__ out)
{
    const int row = blockIdx.x;
    const int t = threadIdx.x;
    __shared__ float red[256];
    float x = y[(size_t)row * DMD + t];
    red[t] = x; __syncthreads();
    for (int s = 128; s > 0; s >>= 1) { if (t < s) red[t] += red[t + s]; __syncthreads(); }
    float mu = red[0] * (1.0f / DMD);
    __syncthreads();
    float d = x - mu;
    red[t] = d * d; __syncthreads();
    for (int s = 128; s > 0; s >>= 1) { if (t < s) red[t] += red[t + s]; __syncthreads(); }
    float var = red[0] * (1.0f / DMD);
    out[(size_t)row * DMD + t] = d * rsqrtf(var + 1e-6f) * g[t] + be[t];
}

// ---------------------------------------------------------------- pooled softmax head
__global__ __launch_bounds__(256)
void head_kernel(const float* __restrict__ h, const float* __restrict__ Wf,
                 const float* __restrict__ bf, float* __restrict__ out)
{
    const int b = blockIdx.x;
    const int t = threadIdx.x;                 // dm channel
    __shared__ float pooled[DMD];
    __shared__ float lg[NCC];
    float sum = 0.f;
    for (int s = 0; s < SS; ++s) sum += h[((size_t)b * SS + s) * DMD + t];
    pooled[t] = sum * (1.0f / SS);
    __syncthreads();
    if (t < NCC) {
        float acc = bf[t];
        for (int dm = 0; dm < DMD; ++dm) acc += pooled[dm] * Wf[dm * NCC + t];
        lg[t] = acc;
    }
    __syncthreads();
    if (t == 0) {
        float m = lg[0];
        for (int c = 1; c < NCC; ++c) m = fmaxf(m, lg[c]);
        float e[NCC]; float se = 0.f;
        for (int c = 0; c < NCC; ++c) { e[c] = __expf(lg[c] - m); se += e[c]; }
        for (int c = 0; c < NCC; ++c) out[b * NCC + c] = e[c] / se;
    }
}

// ---------------------------------------------------------------- launcher
extern "C" void kernel_launch(void* const* d_in, const int* in_sizes, int n_in,
                              void* d_out, int out_size, void* d_ws, size_t ws_size,
                              hipStream_t stream)
{
    const float* x    = (const float*)d_in[0];
    const float* Win  = (const float*)d_in[1];
    const float* bin0 = (const float*)d_in[2];
    const float* Wq   = (const float*)d_in[3];
    const float* bq   = (const float*)d_in[4];
    const float* Wk   = (const float*)d_in[5];
    const float* bk   = (const float*)d_in[6];
    const float* Wv   = (const float*)d_in[7];
    const float* bv   = (const float*)d_in[8];
    const float* Wo   = (const float*)d_in[9];
    const float* bo   = (const float*)d_in[10];
    const float* W1   = (const float*)d_in[11];
    const float* b1   = (const float*)d_in[12];
    const float* W2   = (const float*)d_in[13];
    const float* b2   = (const float*)d_in[14];
    const float* g1   = (const float*)d_in[15];
    const float* be1  = (const float*)d_in[16];
    const float* g2   = (const float*)d_in[17];
    const float* be2  = (const float*)d_in[18];
    const float* Wf   = (const float*)d_in[19];
    const float* bf   = (const float*)d_in[20];
    float* out = (float*)d_out;

    float* w = (float*)d_ws;
    size_t off = 0;
    float* pe = w + off; off += (size_t)SS * DMD;
    float* h  = w + off; off += (size_t)BB * SS * DMD;
    float* qb = w + off; off += (size_t)BB * SS * DMD;
    float* kb = w + off; off += (size_t)BB * SS * DMD;
    float* vb = w + off; off += (size_t)BB * SS * DMD;
    float* ma = w + off; off += (size_t)BB * HH * SS;
    int*   ix = (int*)(w + off); off += 4096;
    float* ob = w + off; off += (size_t)BB * SS * DMD;
    float* yb = w + off; off += (size_t)BB * SS * DMD;
    float* h1 = w + off; off += (size_t)BB * SS * DMD;
    float* tb = w + off; off += (size_t)BB * SS * DFFF;

    const int M = BB * SS;
    const dim3 gblk(32, 4);

    pe_kernel<<<(SS * DMD) / 256, 256, 0, stream>>>(pe);
    // h = x @ Win + bin0 + PE
    gemm_f32_wmma<EPI_PE><<<dim3(DMD / 64, M / 64), gblk, 0, stream>>>(
        x, Win, bin0, pe, h, M, DMD, FFI);

    for (int l = 0; l < NLL; ++l) {
        const float* Wq_l = Wq + (size_t)l * DMD * DMD;
        const float* Wk_l = Wk + (size_t)l * DMD * DMD;
        const float* Wv_l = Wv + (size_t)l * DMD * DMD;
        const float* Wo_l = Wo + (size_t)l * DMD * DMD;
        const float* W1_l = W1 + (size_t)l * DMD * DFFF;
        const float* W2_l = W2 + (size_t)l * DFFF * DMD;
        const float* bq_l = bq + (size_t)l * DMD;
        const float* bk_l = bk + (size_t)l * DMD;
        const float* bv_l = bv + (size_t)l * DMD;
        const float* bo_l = bo + (size_t)l * DMD;
        const float* b1_l = b1 + (size_t)l * DFFF;
        const float* b2_l = b2 + (size_t)l * DMD;
        const float* g1_l = g1 + (size_t)l * DMD;
        const float* be1_l = be1 + (size_t)l * DMD;
        const float* g2_l = g2 + (size_t)l * DMD;
        const float* be2_l = be2 + (size_t)l * DMD;

        // Q/K/V projections, scattered straight into [B,H,S,D]
        gemm_f32_wmma<EPI_HEADS><<<dim3(DMD / 64, M / 64), gblk, 0, stream>>>(
            h, Wq_l, bq_l, nullptr, qb, M, DMD, DMD);
        gemm_f32_wmma<EPI_HEADS><<<dim3(DMD / 64, M / 64), gblk, 0, stream>>>(
            h, Wk_l, bk_l, nullptr, kb, M, DMD, DMD);
        gemm_f32_wmma<EPI_HEADS><<<dim3(DMD / 64, M / 64), gblk, 0, stream>>>(
            h, Wv_l, bv_l, nullptr, vb, M, DMD, DMD);

        // streaming QK^T probe -> per-key exp sums
        hipMemsetAsync(ma, 0, (size_t)BB * HH * SS * sizeof(float), stream);
        probe_kernel<<<dim3(SS / 16, BB * HH), dim3(32, 8), 0, stream>>>(qb, kb, ma);
        topk_kernel<<<BB * HH, 1024, 0, stream>>>(ma, ix);
        sparse_attn_kernel<<<dim3(SS / 256, BB * HH), 256, 0, stream>>>(qb, kb, vb, ix, ob);

        // out projection + residual, LN
        gemm_f32_wmma<EPI_RES><<<dim3(DMD / 64, M / 64), gblk, 0, stream>>>(
            ob, Wo_l, bo_l, h, yb, M, DMD, DMD);
        ln_kernel<<<M, 256, 0, stream>>>(yb, g1_l, be1_l, h1);

        // FFN
        gemm_f32_wmma<EPI_RELU><<<dim3(DFFF / 64, M / 64), gblk, 0, stream>>>(
            h1, W1_l, b1_l, nullptr, tb, M, DFFF, DMD);
        gemm_f32_wmma<EPI_RES><<<dim3(DMD / 64, M / 64), gblk, 0, stream>>>(
            tb, W2_l, b2_l, h1, yb, M, DMD, DFFF);
        ln_kernel<<<M, 256, 0, stream>>>(yb, g2_l, be2_l, h);
    }

    head_kernel<<<BB, 256, 0, stream>>>(h, Wf, bf, out);
}